// GreedyRNNT_84344567759683
// MI455X (gfx1250) — compile-verified
//
#include <hip/hip_runtime.h>
#include <hip/hip_bf16.h>
#include <math.h>

// ---------------------------------------------------------------------------
// Greedy RNN-T decode for MI455X (gfx1250, wave32).
//   Phase 1: enc_proj = encoder_outputs @ We  (parallel WMMA f32 GEMM)
//   Phase 2: persistent decode kernel, 64 blocks == B, device-wide software
//            barrier; all per-step GEMMs via V_WMMA_F32_16X16X4_F32 so the
//            67 MB weight set stays L2-resident and fp32 semantics are kept
//            (greedy argmax is precision-sensitive).
// Workspace need: ~86 MB (enc_proj 82 MB + logits 2.6 MB + small state).
// ---------------------------------------------------------------------------

typedef __attribute__((ext_vector_type(2))) float v2f;
typedef __attribute__((ext_vector_type(8))) float v8f;

namespace {
constexpr int B = 64, T = 500, E = 512, D = 640, J = 640, V = 10000;
constexpr int G4 = 4 * D;                    // 2560
constexpr int MAX_SYM = 3;
constexpr int BLANK = 0;
constexpr int HYP_W = T * MAX_SYM + 1;       // 1501

constexpr int DEC_BLOCKS  = 64;              // must equal B (stage-5 mapping)
constexpr int DEC_THREADS = 256;             // 8 wave32 per block
constexpr int DEC_WAVES   = DEC_BLOCKS * (DEC_THREADS / 32);  // 512 waves
static_assert(DEC_BLOCKS == B, "block<->batch-row mapping");
static_assert(V % 16 == 0 && D % 16 == 0 && J % 16 == 0 && E % 4 == 0, "");
}  // namespace

__device__ __forceinline__ float sigm(float x) { return 1.0f / (1.0f + expf(-x)); }

// Accumulate a 16x16 f32 tile over K using V_WMMA_F32_16X16X4_F32.
//  arow : pointer to this lane's A row (contiguous in k)
//  bcol : pointer to W[0][n0 + (lane&15)]; element (k) at bcol[k*ldb]
// A layout (ISA 7.12.2, 32-bit A 16x4): lanes 0-15 -> K = k0,k0+1;
// lanes 16-31 -> K = k0+2,k0+3.  B mirrors with N striped across lanes.
__device__ __forceinline__ v8f wmma_k_acc(v8f acc,
                                          const float* __restrict__ arow,
                                          const float* __restrict__ bcol,
                                          int ldb, int K) {
  const int lane = threadIdx.x & 31;
  const int koff = (lane & 16) ? 2 : 0;
#pragma unroll 4
  for (int k0 = 0; k0 < K; k0 += 4) {
    const int ka = k0 + koff;
    v2f a, b;
    a.x = arow[ka];
    a.y = arow[ka + 1];
    b.x = bcol[(size_t)ka * ldb];
    b.y = bcol[(size_t)(ka + 1) * ldb];
    acc = __builtin_amdgcn_wmma_f32_16x16x4_f32(false, a, false, b,
                                                (short)0, acc, false, false);
  }
  return acc;
}

// Device-wide barrier (all DEC_BLOCKS resident). Generation counter + reset.
__device__ __forceinline__ void grid_sync(unsigned* cnt, unsigned* gen,
                                          unsigned nblk) {
  __syncthreads();
  __threadfence();
  if (threadIdx.x == 0) {
    volatile unsigned* vgen = gen;
    const unsigned g0 = *vgen;
    if (atomicAdd(cnt, 1u) == nblk - 1u) {
      *cnt = 0u;
      __threadfence();
      atomicAdd(gen, 1u);
    } else {
      while (*vgen == g0) { __builtin_amdgcn_s_sleep(1); }
    }
  }
  __threadfence();
  __syncthreads();
}

// ---------------------------------------------------------------------------
// Phase 1: enc_proj[t][b][j] = sum_e enc[b][t][e] * We[e][j]
// (B,T,E) is a plain 32000x512 row-major A matrix.
// ---------------------------------------------------------------------------
__global__ __launch_bounds__(256) void enc_proj_kernel(
    const float* __restrict__ enc, const float* __restrict__ We,
    float* __restrict__ encp) {
  const int wave   = (blockIdx.x * blockDim.x + threadIdx.x) >> 5;
  const int nwaves = (gridDim.x * blockDim.x) >> 5;
  const int lane = threadIdx.x & 31;
  const int half = lane >> 4, idx = lane & 15;
  constexpr int TM = (B * T) / 16;  // 2000
  constexpr int TN = J / 16;        // 40
  for (int tt = wave; tt < TM * TN; tt += nwaves) {
    const int tm = tt / TN, tn = tt % TN;
    const int n0 = tn * 16;
    const int arow = tm * 16 + idx;
    v8f acc = {};
    acc = wmma_k_acc(acc, enc + (size_t)arow * E, We + n0 + idx, J, E);
#pragma unroll
    for (int r = 0; r < 8; ++r) {
      const int m = tm * 16 + r + 8 * half;   // C layout: VGPR r, half-wave
      const int b = m / T, t = m % T;
      encp[((size_t)t * B + b) * J + (n0 + idx)] = acc[r];
    }
  }
}

// ---------------------------------------------------------------------------
// Phase 2: persistent greedy decode.
// ---------------------------------------------------------------------------
__global__ __launch_bounds__(DEC_THREADS) void rnnt_decode_kernel(
    const float* __restrict__ encp, const int* __restrict__ enc_len,
    const float* __restrict__ embed, const float* __restrict__ Wx,
    const float* __restrict__ Wh, const float* __restrict__ b_lstm,
    const float* __restrict__ Wd, const float* __restrict__ b_j,
    const float* __restrict__ Wv, const float* __restrict__ b_v,
    float* __restrict__ g, float* __restrict__ h, float* __restrict__ c,
    float* __restrict__ h2, float* __restrict__ c2, float* __restrict__ u,
    float* __restrict__ logits, int* __restrict__ cur_tok,
    int* __restrict__ end_flag, unsigned* __restrict__ bar,
    float* __restrict__ out_hyps, float* __restrict__ out_scores) {
  const int tid  = threadIdx.x;
  const int gtid = blockIdx.x * DEC_THREADS + tid;
  const int wave = gtid >> 5;
  const int lane = tid & 31;
  const int half = lane >> 4, idx = lane & 15;
  constexpr int NTHREADS = DEC_BLOCKS * DEC_THREADS;

  unsigned* bar_cnt = bar;
  unsigned* bar_gen = bar + 1;

  __shared__ float smax[DEC_THREADS];
  __shared__ int   sidx[DEC_THREADS];
  __shared__ float ssum[DEC_THREADS];
  __shared__ int   s_eq;

  for (int t = 0; t < T; ++t) {
    for (int s = 0; s < MAX_SYM; ++s) {
      // ---- stage 1: gates g = embed[tok] @ Wx + h @ Wh + b_lstm -----------
      {
        constexpr int TN = G4 / 16;  // 160 -> 640 tiles
        for (int tt = wave; tt < 4 * TN; tt += DEC_WAVES) {
          const int tm = tt / TN, tn = tt % TN;
          const int n0 = tn * 16;
          const int m = tm * 16 + idx;
          const float* xrow = embed + (size_t)cur_tok[m] * D;
          const float* hrow = h + (size_t)m * D;
          v8f acc = {};
          acc = wmma_k_acc(acc, xrow, Wx + n0 + idx, G4, D);
          acc = wmma_k_acc(acc, hrow, Wh + n0 + idx, G4, D);
          const float bias = b_lstm[n0 + idx];
#pragma unroll
          for (int r = 0; r < 8; ++r) {
            const int row = tm * 16 + r + 8 * half;
            g[(size_t)row * G4 + (n0 + idx)] = acc[r] + bias;
          }
        }
      }
      grid_sync(bar_cnt, bar_gen, DEC_BLOCKS);

      // ---- stage 2: LSTM pointwise ---------------------------------------
      for (int i = gtid; i < B * D; i += NTHREADS) {
        const int b = i / D, k = i - b * D;
        const float* gb = g + (size_t)b * G4;
        const float ig = gb[k], fg = gb[k + D];
        const float cg = gb[k + 2 * D], og = gb[k + 3 * D];
        const float c2v = sigm(fg) * c[i] + sigm(ig) * tanhf(cg);
        const float h2v = sigm(og) * tanhf(c2v);
        c2[i] = c2v;
        h2[i] = h2v;
      }
      grid_sync(bar_cnt, bar_gen, DEC_BLOCKS);

      // ---- stage 3: u = tanh(enc_j + h2 @ Wd + b_j) ----------------------
      {
        const float* enct = encp + (size_t)t * B * J;
        constexpr int TN = J / 16;  // 40 -> 160 tiles
        for (int tt = wave; tt < 4 * TN; tt += DEC_WAVES) {
          const int tm = tt / TN, tn = tt % TN;
          const int n0 = tn * 16;
          const int m = tm * 16 + idx;
          v8f acc = {};
          acc = wmma_k_acc(acc, h2 + (size_t)m * D, Wd + n0 + idx, J, D);
#pragma unroll
          for (int r = 0; r < 8; ++r) {
            const int row = tm * 16 + r + 8 * half;
            const int col = n0 + idx;
            u[(size_t)row * J + col] =
                tanhf(acc[r] + enct[(size_t)row * J + col] + b_j[col]);
          }
        }
      }
      grid_sync(bar_cnt, bar_gen, DEC_BLOCKS);

      // ---- stage 4: logits = u @ Wv + b_v --------------------------------
      {
        constexpr int TN = V / 16;  // 625 -> 2500 tiles
        for (int tt = wave; tt < 4 * TN; tt += DEC_WAVES) {
          const int tm = tt / TN, tn = tt % TN;
          const int n0 = tn * 16;
          const int m = tm * 16 + idx;
          v8f acc = {};
          acc = wmma_k_acc(acc, u + (size_t)m * J, Wv + n0 + idx, V, J);
#pragma unroll
          for (int r = 0; r < 8; ++r) {
            const int row = tm * 16 + r + 8 * half;
            const int col = n0 + idx;
            logits[(size_t)row * V + col] = acc[r] + b_v[col];
          }
        }
      }
      grid_sync(bar_cnt, bar_gen, DEC_BLOCKS);

      // ---- stage 5: argmax / log-softmax / state update (block b = row b)
      {
        const int b = blockIdx.x;
        const float* lrow = logits + (size_t)b * V;
        float bm = -INFINITY;
        int bi = 0;
        for (int v = tid; v < V; v += DEC_THREADS) {
          const float lv = lrow[v];
          if (lv > bm) { bm = lv; bi = v; }   // ascending v keeps first max
        }
        smax[tid] = bm; sidx[tid] = bi;
        __syncthreads();
        for (int off = DEC_THREADS / 2; off > 0; off >>= 1) {
          if (tid < off) {
            const float om = smax[tid + off];
            const int   oi = sidx[tid + off];
            if (om > smax[tid] || (om == smax[tid] && oi < sidx[tid])) {
              smax[tid] = om; sidx[tid] = oi;
            }
          }
          __syncthreads();
        }
        const float rowmax = smax[0];
        const int   nt     = sidx[0];
        float ls = 0.0f;
        for (int v = tid; v < V; v += DEC_THREADS) ls += expf(lrow[v] - rowmax);
        ssum[tid] = ls;
        __syncthreads();
        for (int off = DEC_THREADS / 2; off > 0; off >>= 1) {
          if (tid < off) ssum[tid] += ssum[tid + off];
          __syncthreads();
        }
        if (tid == 0) {
          out_scores[b] = -logf(ssum[0]);       // max(log_softmax) = -log(sumexp)
          const int eq = (nt == BLANK) ? 1 : 0;
          const int prev_end = (s == 0) ? 0 : end_flag[b];
          const int endf = prev_end | eq;
          end_flag[b] = endf;
          if (!eq) cur_tok[b] = nt;
          const int lflag = (enc_len[b] <= t) || endf;
          out_hyps[(size_t)b * HYP_W + 1 + t * MAX_SYM + s] =
              (float)(lflag ? BLANK : nt);
          s_eq = eq;
        }
        __syncthreads();
        if (!s_eq) {                            // h,c <- h2,c2 when non-blank
          for (int k = tid; k < D; k += DEC_THREADS) {
            h[(size_t)b * D + k] = h2[(size_t)b * D + k];
            c[(size_t)b * D + k] = c2[(size_t)b * D + k];
          }
        }
      }
      grid_sync(bar_cnt, bar_gen, DEC_BLOCKS);
    }
  }
}

// ---------------------------------------------------------------------------
__global__ void init_kernel(float* h, float* c, int* cur_tok, int* end_flag,
                            unsigned* bar, float* out_hyps) {
  const int i = blockIdx.x * blockDim.x + threadIdx.x;
  if (i < B * D) { h[i] = 0.0f; c[i] = 0.0f; }
  if (i < B) {
    cur_tok[i] = BLANK;
    end_flag[i] = 0;
    out_hyps[(size_t)i * HYP_W] = 0.0f;       // hyps[:,0] = 0
  }
  if (i < 8) bar[i] = 0u;
}

extern "C" void kernel_launch(void* const* d_in, const int* in_sizes, int n_in,
                              void* d_out, int out_size, void* d_ws,
                              size_t ws_size, hipStream_t stream) {
  (void)in_sizes; (void)n_in; (void)out_size; (void)ws_size;
  const float* enc     = (const float*)d_in[0];
  const int*   enc_len = (const int*)d_in[1];
  const float* embed   = (const float*)d_in[2];
  const float* Wx      = (const float*)d_in[3];
  const float* Wh      = (const float*)d_in[4];
  const float* b_lstm  = (const float*)d_in[5];
  const float* We      = (const float*)d_in[6];
  const float* Wd      = (const float*)d_in[7];
  const float* b_j     = (const float*)d_in[8];
  const float* Wv      = (const float*)d_in[9];
  const float* b_v     = (const float*)d_in[10];

  float* out        = (float*)d_out;
  float* out_hyps   = out;                         // B * HYP_W (ints as floats)
  float* out_scores = out + (size_t)B * HYP_W;     // B

  char* p = (char*)d_ws;
  auto take = [&](size_t n) {
    float* r = (float*)p;
    p += n * sizeof(float);
    return r;
  };
  float* encp   = take((size_t)T * B * J);   // 20.48M floats
  float* g      = take((size_t)B * G4);
  float* h      = take((size_t)B * D);
  float* c      = take((size_t)B * D);
  float* h2     = take((size_t)B * D);
  float* c2     = take((size_t)B * D);
  float* u      = take((size_t)B * J);
  float* logits = take((size_t)B * V);
  int*      cur_tok  = (int*)take(B);
  int*      end_flag = (int*)take(B);
  unsigned* bar      = (unsigned*)take(8);

  init_kernel<<<(B * D + 255) / 256, 256, 0, stream>>>(h, c, cur_tok, end_flag,
                                                       bar, out_hyps);
  enc_proj_kernel<<<512, 256, 0, stream>>>(enc, We, encp);
  rnnt_decode_kernel<<<DEC_BLOCKS, DEC_THREADS, 0, stream>>>(
      encp, enc_len, embed, Wx, Wh, b_lstm, Wd, b_j, Wv, b_v, g, h, c, h2, c2,
      u, logits, cur_tok, end_flag, bar, out_hyps, out_scores);
}